// NeurRegLoss_80135499809392
// MI455X (gfx1250) — compile-verified
//
#include <hip/hip_runtime.h>
#include <hip/hip_bf16.h>

typedef float v2f __attribute__((ext_vector_type(2)));
typedef float v8f __attribute__((ext_vector_type(8)));

#define TILE 16
#define HALO 20          // TILE + 2*2 halo
#define OMEGA_F 12582912.0f   // 2*3*128^3
#define OMEGA_I 4194304.0f    // 2*1*128^3

#define AS1 __attribute__((address_space(1)))
#define AS3 __attribute__((address_space(3)))

#if __has_builtin(__builtin_amdgcn_global_load_async_to_lds_b32)
#define USE_ASYNC_LDS 1
#else
#define USE_ASYNC_LDS 0
#endif

__device__ __forceinline__ void wait_asynccnt0() {
#if __has_builtin(__builtin_amdgcn_s_wait_asynccnt)
    __builtin_amdgcn_s_wait_asynccnt(0);
#else
    asm volatile("s_wait_asynccnt 0x0" ::: "memory");
#endif
}

// workspace layout (floats):
// [0]=ssdF  [1]=tv0_prod [2]=tv0_sum [3]=tv1_prod [4]=tv1_sum
// [5..7]=lcc0 {ab,aa,bb}  [8..10]=lcc1 {ab,aa,bb}

__global__ __launch_bounds__(32) void zero_ws(float* ws) {
    if (threadIdx.x < 11) ws[threadIdx.x] = 0.0f;
}

// ---------------- SSD reduction for reg_field_loss ----------------
__global__ __launch_bounds__(256) void ssd_kernel(const float* __restrict__ a,
                                                  const float* __restrict__ b,
                                                  float* __restrict__ ws, int n4) {
    const float4* A = (const float4*)a;
    const float4* B = (const float4*)b;
    float s = 0.0f;
    size_t stride = (size_t)gridDim.x * blockDim.x;
    for (size_t i = (size_t)blockIdx.x * blockDim.x + threadIdx.x; i < (size_t)n4; i += stride) {
        float4 x = A[i], y = B[i];
        float d0 = x.x - y.x, d1 = x.y - y.y, d2 = x.z - y.z, d3 = x.w - y.w;
        s += d0*d0 + d1*d1 + d2*d2 + d3*d3;
    }
    for (int o = 16; o > 0; o >>= 1) s += __shfl_xor(s, o, 32);
    __shared__ float red[8];
    if ((threadIdx.x & 31) == 0) red[threadIdx.x >> 5] = s;
    __syncthreads();
    if (threadIdx.x == 0) {
        float tot = 0.0f;
        #pragma unroll
        for (int i = 0; i < 8; ++i) tot += red[i];
        atomicAdd(&ws[0], tot);
    }
}

// ---------------- Tversky sums ----------------
__global__ __launch_bounds__(256) void tversky_kernel(const float* __restrict__ a,
                                                      const float* __restrict__ b,
                                                      float* __restrict__ ws2, int n4) {
    const float4* A = (const float4*)a;
    const float4* B = (const float4*)b;
    float sp = 0.0f, ss = 0.0f;
    size_t stride = (size_t)gridDim.x * blockDim.x;
    for (size_t i = (size_t)blockIdx.x * blockDim.x + threadIdx.x; i < (size_t)n4; i += stride) {
        float4 x = A[i], y = B[i];
        sp += x.x*y.x + x.y*y.y + x.z*y.z + x.w*y.w;
        ss += x.x+y.x + x.y+y.y + x.z+y.z + x.w+y.w;
    }
    for (int o = 16; o > 0; o >>= 1) { sp += __shfl_xor(sp, o, 32); ss += __shfl_xor(ss, o, 32); }
    __shared__ float rp[8], rs[8];
    if ((threadIdx.x & 31) == 0) { rp[threadIdx.x >> 5] = sp; rs[threadIdx.x >> 5] = ss; }
    __syncthreads();
    if (threadIdx.x == 0) {
        float tp = 0.0f, ts = 0.0f;
        #pragma unroll
        for (int i = 0; i < 8; ++i) { tp += rp[i]; ts += rs[i]; }
        atomicAdd(&ws2[0], tp);
        atomicAdd(&ws2[1], ts);
    }
}

// ---------------- LCC: 5x5x5 box mean (separable) + moment sums ----------------
// grid: (8, 8, 4) -> x-tile, y-tile, z = pair*2 + n.  block: 256 = 8 waves,
// wave w owns output z in [16w, 16w+16). y-tap runs on the matrix pipe as
// P(16x16) = M(16x20) x X(20x16) via five chained V_WMMA_F32_16X16X4_F32.
// Halo planes stream global -> LDS through GLOBAL_LOAD_ASYNC_TO_LDS_B32.
// Outer z loop kept dynamic (4 x unroll-5) to bound VGPR pressure; ring
// indices are compile-time constants within the unrolled body.
__global__ __launch_bounds__(256) void lcc_kernel(const float* __restrict__ I0,
                                                  const float* __restrict__ I0R,
                                                  const float* __restrict__ I1,
                                                  const float* __restrict__ I1R,
                                                  float* __restrict__ ws) {
    const int tid  = threadIdx.x;
    const int wave = tid >> 5;
    const int lane = tid & 31;
    const int hv   = lane >> 4;     // half-wave index
    const int col  = lane & 15;     // N index (x) / M row for A
    const int x0   = blockIdx.x * TILE;
    const int y0   = blockIdx.y * TILE;
    const int n    = blockIdx.z & 1;
    const int pair = blockIdx.z >> 1;
    const float* gt = pair ? I1  : I0;
    const float* pr = pair ? I1R : I0R;
    const int z0 = wave * TILE;

    __shared__ float s_raw[8][2][HALO * HALO]; // 25.6 KB
    __shared__ float s_xf [8][2][HALO * TILE]; // 20.5 KB
    float* raw0 = s_raw[wave][0];
    float* raw1 = s_raw[wave][1];
    float* xf0  = s_xf[wave][0];
    float* xf1  = s_xf[wave][1];

    // z-invariant per-lane element offsets within a plane; -1 = outside tensor.
    int off_k[13];
    #pragma unroll
    for (int k = 0; k < 13; ++k) {
        int e = lane + 32 * k;
        int row = e / HALO, cc = e % HALO;
        int gy = y0 - 2 + row, gx = x0 - 2 + cc;
        bool sp = (e < HALO * HALO) && (unsigned)gy < 128u && (unsigned)gx < 128u;
        off_k[k] = sp ? (gy * 128 + gx) : -1;
    }

    // Constant banded A chunks: M[m][k] = 0.2 if k in [m, m+4].
    // A layout (16x4 f32): lanes 0-15 -> K=0,1 in v0,v1 ; lanes 16-31 -> K=2,3.
    v2f A5[5];
    #pragma unroll
    for (int c = 0; c < 5; ++c) {
        int k0 = 4 * c + 2 * hv;
        int k1 = k0 + 1;
        A5[c].x = (k0 >= col && k0 <= col + 4) ? 0.2f : 0.0f;
        A5[c].y = (k1 >= col && k1 <= col + 4) ? 0.2f : 0.0f;
    }

    // Pre-zero the raw tiles: spatially-OOB slots stay zero forever.
    #pragma unroll
    for (int k = 0; k < 13; ++k) {
        int e = lane + 32 * k;
        if (e < HALO * HALO) { raw0[e] = 0.0f; raw1[e] = 0.0f; }
    }
    __syncthreads();   // DS zero-stores complete before async-pipe writes

    v8f   pring[2][5];          // xy-filtered plane ring (C/D layout), per volume
    float rawc[2][5][8];        // raw center-value ring (lookback 2)
    float s_ab = 0.0f, s_aa = 0.0f, s_bb = 0.0f;

    #pragma unroll 1
    for (int o = 0; o < 4; ++o) {
        #pragma unroll
        for (int u = 0; u < 5; ++u) {
            const int t = 5 * o + u;
            const int j = z0 - 2 + t;              // input plane index
            const bool zin = (j >= 0) && (j < 128);

            // -------- halo-plane load: global -> LDS (async pipe) --------
            if (zin) {
                const size_t jb = ((size_t)(n * 128 + j)) << 14;  // *128*128
                const float* g0 = gt + jb;
                const float* g1 = pr + jb;
                #pragma unroll
                for (int k = 0; k < 13; ++k) {
                    if (off_k[k] >= 0) {
                        int e = lane + 32 * k;
#if USE_ASYNC_LDS
                        __builtin_amdgcn_global_load_async_to_lds_b32(
                            (AS1 int*)(g0 + off_k[k]), (AS3 int*)&raw0[e], 0, 0);
                        __builtin_amdgcn_global_load_async_to_lds_b32(
                            (AS1 int*)(g1 + off_k[k]), (AS3 int*)&raw1[e], 0, 0);
#else
                        raw0[e] = g0[off_k[k]];
                        raw1[e] = g1[off_k[k]];
#endif
                    }
                }
            } else {
                #pragma unroll
                for (int k = 0; k < 13; ++k) {
                    int e = lane + 32 * k;
                    if (e < HALO * HALO) { raw0[e] = 0.0f; raw1[e] = 0.0f; }
                }
            }
#if USE_ASYNC_LDS
            wait_asynccnt0();
#endif
            __syncthreads();

            // -------- x-axis 5-tap (one lane per halo row) --------
            if (lane < HALO) {
                const float* r0 = &raw0[lane * HALO];
                const float* r1 = &raw1[lane * HALO];
                #pragma unroll
                for (int x = 0; x < TILE; ++x) {
                    xf0[lane * TILE + x] = 0.2f * (r0[x] + r0[x+1] + r0[x+2] + r0[x+3] + r0[x+4]);
                    xf1[lane * TILE + x] = 0.2f * (r1[x] + r1[x+1] + r1[x+2] + r1[x+3] + r1[x+4]);
                }
            }
            __syncthreads();

            // -------- y-axis 5-tap on matrix pipe: P = M x X --------
            #pragma unroll
            for (int v = 0; v < 2; ++v) {
                const float* xf  = v ? xf1  : xf0;
                const float* raw = v ? raw1 : raw0;
                v8f acc = {};
                #pragma unroll
                for (int c = 0; c < 5; ++c) {
                    // B layout (4x16 f32): lanes 0-15 -> rows 4c,4c+1 in v0,v1;
                    // lanes 16-31 -> rows 4c+2,4c+3.
                    v2f b;
                    b.x = xf[(4 * c + 2 * hv + 0) * TILE + col];
                    b.y = xf[(4 * c + 2 * hv + 1) * TILE + col];
                    acc = __builtin_amdgcn_wmma_f32_16x16x4_f32(
                            false, A5[c], false, b, (short)0, acc, false, false);
                }
                pring[v][u] = acc;
                // raw center values in C/D layout: (y = r + 8*hv, x = col)
                #pragma unroll
                for (int r = 0; r < 8; ++r)
                    rawc[v][u][r] = raw[(r + 8 * hv + 2) * HALO + (col + 2)];
            }
            __syncthreads();

            // -------- z-axis 5-tap + moment accumulation --------
            // output z = t - 4 + z0; ring holds exactly P(z-2..z+2)
            if (o > 0 || u == 4) {
                v8f mg = (pring[0][0] + pring[0][1] + pring[0][2] + pring[0][3] + pring[0][4]) * 0.2f;
                v8f mp = (pring[1][0] + pring[1][1] + pring[1][2] + pring[1][3] + pring[1][4]) * 0.2f;
                const int s = (u + 3) % 5;  // slot written at t-2 = raw plane of output z
                #pragma unroll
                for (int r = 0; r < 8; ++r) {
                    float dg = rawc[0][s][r] - mg[r];
                    float dp = rawc[1][s][r] - mp[r];
                    s_ab += dg * dp;
                    s_aa += dg * dg;
                    s_bb += dp * dp;
                }
            }
        }
    }

    // wave reduction + global accumulate
    for (int o = 16; o > 0; o >>= 1) {
        s_ab += __shfl_xor(s_ab, o, 32);
        s_aa += __shfl_xor(s_aa, o, 32);
        s_bb += __shfl_xor(s_bb, o, 32);
    }
    if (lane == 0) {
        atomicAdd(&ws[5 + 3 * pair + 0], s_ab);
        atomicAdd(&ws[5 + 3 * pair + 1], s_aa);
        atomicAdd(&ws[5 + 3 * pair + 2], s_bb);
    }
}

// ---------------- final scalar combine ----------------
__global__ __launch_bounds__(32) void final_kernel(const float* __restrict__ ws,
                                                   float* __restrict__ out) {
    if (threadIdx.x != 0) return;
    float reg = sqrtf(ws[0]) / OMEGA_F;
    float ab0 = ws[5], aa0 = ws[6], bb0 = ws[7];
    float ab1 = ws[8], aa1 = ws[9], bb1 = ws[10];
    float l0 = -(ab0 * ab0 / fmaxf(aa0 * bb0, 1e-5f)) / OMEGA_I;
    float l1 = -(ab1 * ab1 / fmaxf(aa1 * bb1, 1e-5f)) / OMEGA_I;
    float tv0 = -ws[1] / fmaxf(ws[2], 1e-5f);
    float tv1 = -ws[3] / fmaxf(ws[4], 1e-5f);
    out[0] = reg + 10.0f * (l0 + l1) + 10.0f * (tv0 + tv1);
}

extern "C" void kernel_launch(void* const* d_in, const int* in_sizes, int n_in,
                              void* d_out, int out_size, void* d_ws, size_t ws_size,
                              hipStream_t stream) {
    const float* F0  = (const float*)d_in[0];
    const float* F0g = (const float*)d_in[1];
    const float* I0  = (const float*)d_in[2];
    const float* I0R = (const float*)d_in[3];
    const float* I1  = (const float*)d_in[4];
    const float* I1R = (const float*)d_in[5];
    const float* S0  = (const float*)d_in[6];
    const float* S0g = (const float*)d_in[7];
    const float* S1  = (const float*)d_in[8];
    const float* S1g = (const float*)d_in[9];
    float* ws  = (float*)d_ws;
    float* out = (float*)d_out;

    zero_ws<<<1, 32, 0, stream>>>(ws);
    ssd_kernel<<<2048, 256, 0, stream>>>(F0, F0g, ws, in_sizes[0] / 4);
    tversky_kernel<<<1024, 256, 0, stream>>>(S0, S0g, ws + 1, in_sizes[6] / 4);
    tversky_kernel<<<1024, 256, 0, stream>>>(S1, S1g, ws + 3, in_sizes[8] / 4);
    lcc_kernel<<<dim3(8, 8, 4), 256, 0, stream>>>(I0, I0R, I1, I1R, ws);
    final_kernel<<<1, 32, 0, stream>>>(ws, out);
}